// HoughRouting1_27857157881939
// MI455X (gfx1250) — compile-verified
//
#include <hip/hip_runtime.h>
#include <hip/hip_bf16.h>

typedef __attribute__((ext_vector_type(2))) float v2f;
typedef __attribute__((ext_vector_type(8))) float v8f;

#define HDIM 512
#define WDIM 1024
#define HW   (HDIM * WDIM)
#define TOPK 200
#define NCAND 208          // 13 tiles of 16
#define NTILES 13
#define CAP 32768          // max survivors kept
#define CENTER_THRESH 3.8917807f   // sum(sqrt(dy^2+dx^2), dy^2+dx^2<=25)/81 + 0.5

// ---- workspace layout (bytes) ----
#define OFF_VOTE 0
#define OFF_SVAL (HW * 4)                    // 2097152
#define OFF_SIDX (OFF_SVAL + CAP * 4)        // 2228224
#define OFF_CNT  (OFF_SIDX + CAP * 4)        // 2359296
#define OFF_ANYV (OFF_CNT + 16)
// packed candidate table: [0..207]=2*cx, [208..415]=2*cy, [416..623]=-(cx^2+cy^2)
#define OFF_CAND (OFF_ANYV + 16)

__global__ void init_kernel(int* cnt) {
    if (threadIdx.x == 0) *cnt = 0;
}

// ---- 11x11 circular-mask Hough vote, LDS-staged 32x32 tile + halo 5 ----
__global__ void vote_kernel(const float* __restrict__ reg, float* __restrict__ vote) {
    __shared__ float sRX[42 * 43];
    __shared__ float sRY[42 * 43];
    const int tx0 = blockIdx.x * 32;
    const int ty0 = blockIdx.y * 32;
    for (int i = threadIdx.x; i < 42 * 42; i += blockDim.x) {
        int ly = i / 42, lx = i - ly * 42;
        int gy = ty0 + ly - 5, gx = tx0 + lx - 5;
        float rx = 0.0f, ry = 0.0f;
        if (gx >= 0 && gx < WDIM && gy >= 0 && gy < HDIM) {
            int g = gy * WDIM + gx;
            rx = reg[g];
            ry = reg[HW + g];
        }
        sRX[ly * 43 + lx] = rx;
        sRY[ly * 43 + lx] = ry;
    }
    __syncthreads();
    const int lx  = threadIdx.x & 31;
    const int ly0 = threadIdx.x >> 5;           // 0..7
    for (int j = 0; j < 4; ++j) {
        const int ly = ly0 + 8 * j;
        float err = 0.0f;
        for (int dy = -5; dy <= 5; ++dy) {
            const int ady = dy < 0 ? -dy : dy;
            const int lim = (ady == 5) ? 0 : (ady == 4) ? 3 : (ady == 0) ? 5 : 4;
            const int rb  = (ly + 5 + dy) * 43 + (lx + 5);
            const float fdy = (float)dy;
            for (int dx = -lim; dx <= lim; ++dx) {
                float ex = (float)dx - sRX[rb + dx];
                float ey = fdy      - sRY[rb + dx];
                err += sqrtf(ex * ex + ey * ey);
            }
        }
        vote[(ty0 + ly) * WDIM + tx0 + lx] = err * (1.0f / 81.0f) + 1.0f;
    }
}

// ---- 7x7 min-pool NMS + threshold; append survivors ----
__global__ void nms_kernel(const float* __restrict__ vote, float* __restrict__ svals,
                           int* __restrict__ sidx, int* __restrict__ cnt) {
    int pix = blockIdx.x * blockDim.x + threadIdx.x;
    if (pix >= HW) return;
    int y = pix >> 10, x = pix & 1023;
    float v = vote[pix];
    float mn = v;
    for (int dy = -3; dy <= 3; ++dy) {
        int yy = y + dy;
        if (yy < 0 || yy >= HDIM) continue;
        for (int dx = -3; dx <= 3; ++dx) {
            int xx = x + dx;
            if (xx < 0 || xx >= WDIM) continue;
            mn = fminf(mn, vote[yy * WDIM + xx]);
        }
    }
    if (mn == v && v < CENTER_THRESH) {
        int s = atomicAdd(cnt, 1);
        if (s < CAP) { svals[s] = v; sidx[s] = pix; }
    }
}

// ---- single-block top-200 by rank counting; build padded candidate table ----
__global__ void topk_kernel(const float* __restrict__ svals, const int* __restrict__ sidx,
                            const int* __restrict__ cnt, float* __restrict__ cand,
                            int* __restrict__ coords, float* __restrict__ cerr,
                            int* __restrict__ anyv) {
    int n = *cnt;
    if (n > CAP) n = CAP;
    for (int k = threadIdx.x; k < NCAND; k += blockDim.x) {
        cand[k] = 0.0f;                  // 2*cx
        cand[NCAND + k] = 0.0f;          // 2*cy
        cand[2 * NCAND + k] = -1.0e30f;  // -(cx^2+cy^2): invalid sentinel
        if (k < TOPK) { coords[2 * k] = 0; coords[2 * k + 1] = 0; cerr[k] = 0.0f; }
    }
    if (threadIdx.x == 0) *anyv = (n > 0) ? 1 : 0;
    __syncthreads();
    for (int i = threadIdx.x; i < n; i += blockDim.x) {
        float v = svals[i];
        int   id = sidx[i];
        int rank = 0;
        for (int j = 0; j < n; ++j) {
            float vj = svals[j];
            int   idj = sidx[j];
            rank += (vj < v) || (vj == v && idj < id);
        }
        if (rank < TOPK) {
            int cy = id >> 10, cx = id & 1023;
            coords[2 * rank]     = cy;
            coords[2 * rank + 1] = cx;
            cerr[rank] = v;
            float fx = (float)cx, fy = (float)cy;
            cand[rank]             = 2.0f * fx;
            cand[NCAND + rank]     = 2.0f * fy;
            cand[2 * NCAND + rank] = -(fx * fx + fy * fy);
        }
    }
}

// ---- WMMA nearest-candidate assignment ----
// score[m][n] = 2*p_m . c_n - |c_n|^2  via  A(16x4) x B(4x16), f32 WMMA.
// A row m = (px, py, 1, 0) ; B col n = (2cx, 2cy, -|c|^2, 0).
// argmax(score) == argmin(dist^2).
// All 13 B tiles are preloaded branchlessly into registers, then 13 WMMAs issue
// back-to-back with the argmax VALU chains interleaved (XDL/VALU co-execution).
__global__ void assign_kernel(const float* __restrict__ fg, const float* __restrict__ reg,
                              const float* __restrict__ cand, const int* __restrict__ anyv,
                              int* __restrict__ inst) {
    const int wave = (blockIdx.x * blockDim.x + threadIdx.x) >> 5;   // 16-pixel group id
    const int lane = threadIdx.x & 31;
    const int half = lane >> 4;
    const int l16  = lane & 15;
    const int base = wave * 16;

    // A operand: lanes 0-15 carry K={0,1} = (px,py); lanes 16-31 carry K={2,3} = (1,0)
    float ax, ay;
    if (!half) {
        const int pix = base + l16;
        const int x = pix & 1023, y = pix >> 10;
        ax = (float)(x + 1) - reg[pix];
        ay = (float)(y + 1) - reg[HW + pix];
    } else {
        ax = 1.0f; ay = 0.0f;
    }
    v2f a; a.x = ax; a.y = ay;

    // Branchless B preload: lanes<16 read row block {2cx | 2cy}; lanes>=16 read
    // row block {-|c|^2 | 0}. Index select, single unconditional load per element.
    const int xoff = half ? (2 * NCAND) : 0;   // K={0 or 2} row source
    float bxv[NTILES], byv[NTILES];
#pragma unroll
    for (int t = 0; t < NTILES; ++t) {
        const int n = t * 16 + l16;
        bxv[t] = cand[xoff + n];
        float cy2 = cand[NCAND + n];           // loaded by all lanes; tiny L2-hot table
        byv[t] = half ? 0.0f : cy2;            // K={1}=2cy, K={3}=0
    }

    float best[8];
    int   bidx[8];
#pragma unroll
    for (int v = 0; v < 8; ++v) { best[v] = -3.4e38f; bidx[v] = 0; }

#pragma unroll
    for (int t = 0; t < NTILES; ++t) {
        v2f b; b.x = bxv[t]; b.y = byv[t];
        v8f c = {};
        v8f d = __builtin_amdgcn_wmma_f32_16x16x4_f32(
            false, a, false, b, (short)0, c, false, false);
        const int n0 = t * 16 + l16;           // candidate column this lane owns in D
#pragma unroll
        for (int v = 0; v < 8; ++v) {
            float s = d[v];
            if (s > best[v]) { best[v] = s; bidx[v] = n0; }
        }
    }

    // reduce over N (16 lanes within each half); xor masks < 16 stay in-half
#pragma unroll
    for (int m = 1; m <= 8; m <<= 1) {
#pragma unroll
        for (int v = 0; v < 8; ++v) {
            float ov = __shfl_xor(best[v], m, 32);
            int   oi = __shfl_xor(bidx[v], m, 32);
            if (ov > best[v] || (ov == best[v] && oi < bidx[v])) {
                best[v] = ov; bidx[v] = oi;
            }
        }
    }

    if (l16 == 0) {
        const int av = *anyv;
        const int mbase = half * 8;    // lane0 -> M 0..7, lane16 -> M 8..15
#pragma unroll
        for (int v = 0; v < 8; ++v) {
            const int pix = base + mbase + v;
            const int thing = (fg[pix] >= 0.5f) ? 1 : 0;
            inst[pix] = (thing & av) ? (bidx[v] + 1) : 0;
        }
    }
}

extern "C" void kernel_launch(void* const* d_in, const int* in_sizes, int n_in,
                              void* d_out, int out_size, void* d_ws, size_t ws_size,
                              hipStream_t stream) {
    (void)in_sizes; (void)n_in; (void)out_size; (void)ws_size;
    const float* fg  = (const float*)d_in[0];   // (1,1,H,W)
    const float* reg = (const float*)d_in[1];   // (1,2,H,W)

    char* ws = (char*)d_ws;
    float* vote  = (float*)(ws + OFF_VOTE);
    float* svals = (float*)(ws + OFF_SVAL);
    int*   sidx  = (int*)  (ws + OFF_SIDX);
    int*   cnt   = (int*)  (ws + OFF_CNT);
    int*   anyv  = (int*)  (ws + OFF_ANYV);
    float* cand  = (float*)(ws + OFF_CAND);

    char* ob = (char*)d_out;
    int*   inst   = (int*)  ob;                      // (1,H,W) int32
    int*   coords = (int*)  (ob + (size_t)HW * 4);   // (1,200,2) int32
    float* cerr   = (float*)(ob + (size_t)(HW + 2 * TOPK) * 4); // (1,200) f32

    init_kernel<<<1, 32, 0, stream>>>(cnt);
    {
        dim3 g(WDIM / 32, HDIM / 32);
        vote_kernel<<<g, 256, 0, stream>>>(reg, vote);
    }
    nms_kernel<<<HW / 256, 256, 0, stream>>>(vote, svals, sidx, cnt);
    topk_kernel<<<1, 256, 0, stream>>>(svals, sidx, cnt, cand, coords, cerr, anyv);
    {
        // 32768 pixel-groups of 16; 8 waves/block => 4096 blocks, all waves full
        assign_kernel<<<(HW / 16) / 8, 256, 0, stream>>>(fg, reg, cand, anyv, inst);
    }
}